// SGALayer_82995948028057
// MI455X (gfx1250) — compile-verified
//
#include <hip/hip_runtime.h>
#include <math.h>

typedef __attribute__((ext_vector_type(2))) float v2f;
typedef __attribute__((ext_vector_type(8))) float v8f;

#define NROWS 100000
#define D 256
#define DP 260                 // padded LDS row stride (floats); 260*4=1040B, 16B-aligned rows
#define ROW_TILES (NROWS / 16) // 6250
#define KV_SLABS 50
#define SLAB_ROWS 2000         // NROWS / KV_SLABS

#define USE_ASYNC_LDS 1

// ---------------------------------------------------------------------------
// Async Global -> LDS staging helpers (CDNA5 GLOBAL_LOAD_ASYNC_TO_LDS_B128,
// tracked by ASYNCcnt). LDS offset = ptrtoint of flat->as3 addrspacecast.
// ---------------------------------------------------------------------------
typedef __attribute__((address_space(3))) unsigned char lds_byte;

__device__ __forceinline__ unsigned lds_off_u32(void* p) {
    return (unsigned)(size_t)(lds_byte*)p;
}

__device__ __forceinline__ void stage16(const float* __restrict__ g, float* l) {
#if USE_ASYNC_LDS
    unsigned lo = lds_off_u32(l);
    asm volatile("global_load_async_to_lds_b128 %0, %1, off"
                 :: "v"(lo), "v"(g) : "memory");
#else
    float4 val = *(const float4*)g;
    l[0] = val.x; l[1] = val.y; l[2] = val.z; l[3] = val.w;
#endif
}

__device__ __forceinline__ void stage_wait() {
#if USE_ASYNC_LDS
    asm volatile("s_wait_asynccnt 0x0" ::: "memory");
#endif
}

// ---------------------------------------------------------------------------
// Kernel 1: q/k/v = x @ W + b.  grid=(6250, 3), block=512 (16 waves).
// Each wave computes one 16x16 tile of the 16x256 row-slab via f32 WMMA.
// Epilogue fuses deterministic partials: per-block ksum columns, sum(q^2),
// sum(k^2)  (saves a full re-read of q and k from HBM).
// ---------------------------------------------------------------------------
__global__ __launch_bounds__(512)
void qkv_gemm(const float* __restrict__ x_src, const float* __restrict__ x_dst,
              const float* __restrict__ Wq, const float* __restrict__ bq,
              const float* __restrict__ Wk, const float* __restrict__ bk,
              const float* __restrict__ Wv, const float* __restrict__ bv,
              float* __restrict__ q, float* __restrict__ k, float* __restrict__ v,
              float* __restrict__ pcolk,   // [6250][256] per-block k column sums
              float* __restrict__ pq2blk,  // [6250]
              float* __restrict__ pk2blk)  // [6250]
{
    __shared__ float xs[16 * DP];
    __shared__ float wsq[16];

    const float* X; const float* W; const float* B; float* O;
    if (blockIdx.y == 0)      { X = x_src; W = Wq; B = bq; O = q; }
    else if (blockIdx.y == 1) { X = x_dst; W = Wk; B = bk; O = k; }
    else                      { X = x_dst; W = Wv; B = bv; O = v; }

    const int r0  = blockIdx.x * 16;
    const int tid = threadIdx.x;

    // Stage 16x256 x-tile into padded LDS (1024 x 16B, async copy engine)
    for (int i = tid; i < 16 * (D / 4); i += 512) {
        const int row = i >> 6;          // 64 float4 per row
        const int c4  = i & 63;
        stage16(X + (size_t)(r0 + row) * D + c4 * 4, &xs[row * DP + c4 * 4]);
    }
    stage_wait();
    __syncthreads();

    const int wave  = tid >> 5;          // 0..15 -> column tile
    const int lane  = tid & 31;
    const int n0    = wave * 16;
    const int m16   = lane & 15;
    const int khalf = (lane >> 4) * 2;   // lanes 0-15: K 0,1 ; lanes 16-31: K 2,3

    v8f c = {};
    const float* ap = &xs[m16 * DP + khalf];
    const float* bp = W + (size_t)khalf * D + n0 + m16;
    #pragma unroll 4
    for (int k0 = 0; k0 < D; k0 += 4) {
        v2f a, b;
        a.x = ap[0]; a.y = ap[1];        // A[m][k], A[m][k+1]
        b.x = bp[0]; b.y = bp[D];        // B[k][n], B[k+1][n]
        c = __builtin_amdgcn_wmma_f32_16x16x4_f32(false, a, false, b,
                                                  (short)0, c, false, false);
        ap += 4;
        bp += (size_t)4 * D;
    }

    const float bias = B[n0 + m16];
    float* op = O + (size_t)(r0 + (lane >> 4) * 8) * D + n0 + m16;
    float colpart = 0.f, sq = 0.f;
    #pragma unroll
    for (int j = 0; j < 8; ++j) {
        const float val = c[j] + bias;
        op[(size_t)j * D] = val;
        colpart += val;
        sq += val * val;
    }

    if (blockIdx.y == 1) {
        // column sums of k: lane l holds rows 0-7, lane l+16 rows 8-15 of col n0+l
        const float other = __shfl_down(colpart, 16);
        if (lane < 16)
            pcolk[(size_t)blockIdx.x * D + n0 + lane] = colpart + other;
    }
    if (blockIdx.y != 2) {
        // wave-level sum of squares (fixed xor tree -> deterministic)
        #pragma unroll
        for (int o = 16; o > 0; o >>= 1) sq += __shfl_xor(sq, o);
        if (lane == 0) wsq[wave] = sq;
        __syncthreads();
        if (tid == 0) {
            float t = 0.f;
            #pragma unroll
            for (int w = 0; w < 16; ++w) t += wsq[w];
            if (blockIdx.y == 0) pq2blk[blockIdx.x] = t;
            else                 pk2blk[blockIdx.x] = t;
        }
    }
}

// ---------------------------------------------------------------------------
// Kernel 2: combine per-block partials -> ksum[256], s = 1/(||q|| * ||k||)
// ---------------------------------------------------------------------------
__global__ __launch_bounds__(256)
void reduce_scalars(const float* __restrict__ pcolk,
                    const float* __restrict__ pq2blk,
                    const float* __restrict__ pk2blk,
                    float* __restrict__ ksum, float* __restrict__ scal)
{
    __shared__ float sq[256];
    __shared__ float sk[256];
    const int t = threadIdx.x;

    float s_ = 0.f;
    for (int b = 0; b < ROW_TILES; ++b)
        s_ += pcolk[(size_t)b * D + t];
    ksum[t] = s_;

    float q2 = 0.f, k2 = 0.f;
    for (int i = t; i < ROW_TILES; i += 256) {
        q2 += pq2blk[i];
        k2 += pk2blk[i];
    }
    sq[t] = q2; sk[t] = k2;
    __syncthreads();
    for (int off = 128; off > 0; off >>= 1) {
        if (t < off) { sq[t] += sq[t + off]; sk[t] += sk[t + off]; }
        __syncthreads();
    }
    if (t == 0)
        scal[0] = 1.0f / (sqrtf(sq[0]) * sqrtf(sk[0]));
}

// ---------------------------------------------------------------------------
// Kernel 3: split-K partials of kv = k^T v.  grid=(16,16,KV_SLABS), block=32.
// ---------------------------------------------------------------------------
__global__ __launch_bounds__(32)
void kv_partial(const float* __restrict__ k, const float* __restrict__ v,
                float* __restrict__ pkv)
{
    const int tj    = blockIdx.x;        // kv row tile (k-column block)
    const int tn    = blockIdx.y;        // kv col tile (v-column block)
    const int slab  = blockIdx.z;
    const int lane  = threadIdx.x;
    const int m16   = lane & 15;
    const int khalf = (lane >> 4) * 2;
    const size_t i0 = (size_t)slab * SLAB_ROWS;

    const float* kp = k + (i0 + khalf) * D + tj * 16 + m16;
    const float* vp = v + (i0 + khalf) * D + tn * 16 + m16;
    v8f c = {};
    for (int it = 0; it < SLAB_ROWS; it += 4) {
        v2f a, b;
        a.x = kp[0]; a.y = kp[D];        // A[m][kk] = k[i][m]
        b.x = vp[0]; b.y = vp[D];        // B[kk][n] = v[i][n]
        c = __builtin_amdgcn_wmma_f32_16x16x4_f32(false, a, false, b,
                                                  (short)0, c, false, false);
        kp += (size_t)4 * D;
        vp += (size_t)4 * D;
    }
    float* op = pkv + (size_t)slab * D * D
                    + (size_t)(tj * 16 + (lane >> 4) * 8) * D + tn * 16 + m16;
    #pragma unroll
    for (int j = 0; j < 8; ++j)
        op[(size_t)j * D] = c[j];
}

// ---------------------------------------------------------------------------
// Kernel 4: fixed-order reduce of kv partials
// ---------------------------------------------------------------------------
__global__ __launch_bounds__(256)
void kv_reduce(const float* __restrict__ pkv, float* __restrict__ kvm)
{
    const int idx = blockIdx.x * 256 + threadIdx.x;   // 0..65535
    float s = 0.f;
    for (int z = 0; z < KV_SLABS; ++z)
        s += pkv[(size_t)z * D * D + idx];
    kvm[idx] = s;
}

// ---------------------------------------------------------------------------
// Kernel 5: out = (q @ kv * s + N*v) / (q . ksum * s + N).
// ---------------------------------------------------------------------------
__global__ __launch_bounds__(512)
void final_gemm(const float* __restrict__ q, const float* __restrict__ v,
                const float* __restrict__ kvm, const float* __restrict__ ksum,
                const float* __restrict__ scal, float* __restrict__ out)
{
    __shared__ float qs[16 * DP];
    __shared__ float dpart[16 * 32];
    __shared__ float dotv[16];

    const int r0  = blockIdx.x * 16;
    const int tid = threadIdx.x;

    for (int i = tid; i < 16 * (D / 4); i += 512) {
        const int row = i >> 6;
        const int c4  = i & 63;
        stage16(q + (size_t)(r0 + row) * D + c4 * 4, &qs[row * DP + c4 * 4]);
    }
    stage_wait();
    __syncthreads();

    // per-row dot(q_row, ksum): 16 rows x 32 chunks of 8, fixed-order combine
    {
        const int row = tid & 15;
        const int cix = tid >> 4;        // 0..31
        float d = 0.f;
        #pragma unroll
        for (int e = 0; e < 8; ++e)
            d += qs[row * DP + cix * 8 + e] * ksum[cix * 8 + e];
        dpart[row * 32 + cix] = d;
    }
    __syncthreads();
    if (tid < 16) {
        float d = 0.f;
        #pragma unroll
        for (int cix = 0; cix < 32; ++cix) d += dpart[tid * 32 + cix];
        dotv[tid] = d;
    }
    __syncthreads();

    const int wave  = tid >> 5;
    const int lane  = tid & 31;
    const int n0    = wave * 16;
    const int m16   = lane & 15;
    const int khalf = (lane >> 4) * 2;

    v8f c = {};
    const float* ap = &qs[m16 * DP + khalf];
    const float* bp = kvm + (size_t)khalf * D + n0 + m16;
    #pragma unroll 4
    for (int k0 = 0; k0 < D; k0 += 4) {
        v2f a, b;
        a.x = ap[0]; a.y = ap[1];
        b.x = bp[0]; b.y = bp[D];
        c = __builtin_amdgcn_wmma_f32_16x16x4_f32(false, a, false, b,
                                                  (short)0, c, false, false);
        ap += 4;
        bp += (size_t)4 * D;
    }

    const float s  = scal[0];
    const float nn = (float)NROWS;
    const int mrow0 = (lane >> 4) * 8;
    const float* vp = v   + (size_t)(r0 + mrow0) * D + n0 + m16;
    float*       op = out + (size_t)(r0 + mrow0) * D + n0 + m16;
    #pragma unroll
    for (int j = 0; j < 8; ++j) {
        const float down = dotv[mrow0 + j] * s + nn;
        const float up   = c[j] * s + nn * vp[(size_t)j * D];
        op[(size_t)j * D] = up / down;
    }
}

// ---------------------------------------------------------------------------
extern "C" void kernel_launch(void* const* d_in, const int* in_sizes, int n_in,
                              void* d_out, int out_size, void* d_ws, size_t ws_size,
                              hipStream_t stream)
{
    const float* x_src = (const float*)d_in[0];
    const float* x_dst = (const float*)d_in[1];
    const float* Wq    = (const float*)d_in[2];
    const float* bq    = (const float*)d_in[3];
    const float* Wk    = (const float*)d_in[4];
    const float* bk    = (const float*)d_in[5];
    const float* Wv    = (const float*)d_in[6];
    const float* bv    = (const float*)d_in[7];
    float* out = (float*)d_out;
    float* ws  = (float*)d_ws;

    size_t off = 0;
    float* q      = ws + off; off += (size_t)NROWS * D;
    float* k      = ws + off; off += (size_t)NROWS * D;
    float* v      = ws + off; off += (size_t)NROWS * D;
    float* pkv    = ws + off; off += (size_t)KV_SLABS * D * D;
    float* kvm    = ws + off; off += (size_t)D * D;
    float* pcolk  = ws + off; off += (size_t)ROW_TILES * D;
    float* pq2blk = ws + off; off += (size_t)ROW_TILES;
    float* pk2blk = ws + off; off += (size_t)ROW_TILES;
    float* ksum   = ws + off; off += D;
    float* scal   = ws + off; off += 1;

    dim3 g1(ROW_TILES, 3);
    qkv_gemm<<<g1, 512, 0, stream>>>(x_src, x_dst, Wq, bq, Wk, bk, Wv, bv,
                                     q, k, v, pcolk, pq2blk, pk2blk);

    reduce_scalars<<<1, 256, 0, stream>>>(pcolk, pq2blk, pk2blk, ksum, scal);

    dim3 g3(16, 16, KV_SLABS);
    kv_partial<<<g3, 32, 0, stream>>>(k, v, pkv);
    kv_reduce<<<(D * D) / 256, 256, 0, stream>>>(pkv, kvm);

    final_gemm<<<NROWS / 16, 512, 0, stream>>>(q, v, kvm, ksum, scal, out);
}